// SparseNet_40338332844930
// MI455X (gfx1250) — compile-verified
//
#include <hip/hip_runtime.h>
#include <cstdint>

#define NATOMS   128   // dictionary atoms
#define KSZ      16
#define STRIDE   8
#define CDIM     15    // code spatial dim
#define NPOS     225   // CDIM*CDIM
#define NPOS_PAD 240   // padded to 15 tiles of 16
#define PQ       256   // KSZ*KSZ
#define IMG      128
#define BATCH    128
#define MAX_ITER 20
#define R_LR     0.1f
#define LMDA     0.005f

typedef __attribute__((ext_vector_type(2))) float v2f;
typedef __attribute__((ext_vector_type(8))) float v8f;

static __device__ __forceinline__ v8f wmma4(v2f a, v2f b, v8f c) {
    // D = A(16x4 f32) * B(4x16 f32) + C(16x16 f32)
    return __builtin_amdgcn_wmma_f32_16x16x4_f32(
        /*neg_a=*/false, a, /*neg_b=*/false, b,
        /*c_mod=*/(short)0, c, /*reuse_a=*/false, /*reuse_b=*/false);
}

// Async global->LDS staging (CDNA5): per-lane LDS dest + saddr/voffset global.
// INST_OFFSET is added to BOTH the LDS and global addresses (ISA 08 §4.4).
static __device__ __forceinline__ void async_g2l_b128x2(uint32_t lds, uint32_t goff,
                                                        const void* sbase) {
    asm volatile(
        "global_load_async_to_lds_b128 %0, %1, %2\n\t"
        "global_load_async_to_lds_b128 %0, %1, %2 offset:16"
        :: "v"(lds), "v"(goff), "s"((uint64_t)(uintptr_t)sbase) : "memory");
}
static __device__ __forceinline__ void async_g2l_b128x4(uint32_t lds, uint32_t goff,
                                                        const void* sbase) {
    asm volatile(
        "global_load_async_to_lds_b128 %0, %1, %2\n\t"
        "global_load_async_to_lds_b128 %0, %1, %2 offset:16\n\t"
        "global_load_async_to_lds_b128 %0, %1, %2 offset:32\n\t"
        "global_load_async_to_lds_b128 %0, %1, %2 offset:48"
        :: "v"(lds), "v"(goff), "s"((uint64_t)(uintptr_t)sbase) : "memory");
}
static __device__ __forceinline__ void wait_async0() {
    asm volatile("s_wait_asynccnt 0x0" ::: "memory");
}
static __device__ __forceinline__ uint32_t lds_addr(const void* p) {
    return (uint32_t)(uintptr_t)p;   // low 32 bits of generic shared ptr = LDS offset
}

// ---------------- utility kernels ----------------

__global__ void zero_f32(float* __restrict__ p, int n) {
    for (int i = blockIdx.x * blockDim.x + threadIdx.x; i < n;
         i += gridDim.x * blockDim.x)
        p[i] = 0.0f;
}

__global__ void init_ctrl(float* acc, int* flag) {
    acc[0] = 0.0f; acc[1] = 0.0f; *flag = 0;
}

// Pack W into WMMA B-fragment pair layouts:
//  Wd  (decode): Wd2[kp*PQ + pq]     = { W[2kp][pq],  W[2kp+1][pq]  }   kp over atoms/2
//  Wte (encode): Wte2[kp*NATOMS + n] = { W[n][2kp],   W[n][2kp+1]   }   kp over taps/2
__global__ void pack_W(const float* __restrict__ W, float* __restrict__ Wd,
                       float* __restrict__ Wte) {
    int t = blockIdx.x * blockDim.x + threadIdx.x;     // 0..16383
    if (t >= 16384) return;
    {   int kp = t >> 8, pq = t & 255;                 // kp 0..63
        Wd[2 * t + 0] = W[(2 * kp) * PQ + pq];
        Wd[2 * t + 1] = W[(2 * kp + 1) * PQ + pq];
    }
    {   int kp = t >> 7, n = t & 127;                  // kp 0..127
        Wte[2 * t + 0] = W[n * PQ + 2 * kp];
        Wte[2 * t + 1] = W[n * PQ + 2 * kp + 1];
    }
}

__global__ void copy_img(const float4* __restrict__ img, float4* __restrict__ E,
                         int n4, const int* __restrict__ flag) {
    if (*flag) return;
    for (int i = blockIdx.x * blockDim.x + threadIdx.x; i < n4;
         i += gridDim.x * blockDim.x)
        E[i] = img[i];
}

__global__ void check_conv(float* acc, int* flag) {
    float dn = acc[0], dr = acc[1];
    if (dn < 1e-4f * dr) *flag = 1;   // ||Rn-R||/||R|| < 0.01
    acc[0] = 0.0f; acc[1] = 0.0f;
}

// ---------------- decode: dst += scale * col2im(R x W) ----------------
// grid: (15 ij-tiles, 128 batches), 256 threads = 8 waves.
// Wave w computes the 16(ij) x 32(pq) slab [pq tiles 2w, 2w+1], K = atoms.
__global__ __launch_bounds__(256)
void decode_scatter(const float* __restrict__ R, const float* __restrict__ Wd,
                    float* __restrict__ dst, float scale,
                    const int* __restrict__ flag) {
    if (flag && *flag) return;
    const int ijt = blockIdx.x, b = blockIdx.y;
    const int ij0 = ijt * 16;

    __shared__ float As[16 * NATOMS];   // 8 KB A-tile (R rows; padded rows are 0)
    const float* Rb = R + (size_t)b * NPOS_PAD * NATOMS;
    {   // async-stage 32B per thread
        int t = threadIdx.x;
        int r = t >> 4, ch = t & 15;
        async_g2l_b128x2(lds_addr(&As[r * NATOMS + ch * 8]),
                         (uint32_t)(((ij0 + r) * NATOMS + ch * 8) * 4), Rb);
    }
    wait_async0();
    __syncthreads();

    const int lane = threadIdx.x & 31;
    const int w    = threadIdx.x >> 5;
    const int row  = lane & 15;
    const int hi   = lane >> 4;
    const int pq0  = w * 32;
    const v2f* Wd2 = (const v2f*)Wd;

    v8f c0 = {}; v8f c1 = {};
    for (int k = 0; k < NATOMS; k += 4) {
        const int kk = k + 2 * hi;
        const int kp = (k >> 1) + hi;
        v2f a  = *(const v2f*)&As[row * NATOMS + kk];
        v2f b0 = Wd2[kp * PQ + pq0 + row];
        v2f b1 = Wd2[kp * PQ + pq0 + 16 + row];
        c0 = wmma4(a, b0, c0);
        c1 = wmma4(a, b1, c1);
    }

    // col2im overlap-add epilogue
    float* D = dst + (size_t)b * IMG * IMG;
    for (int r = 0; r < 8; ++r) {
        const int m  = r + hi * 8;
        const int ij = ij0 + m;
        if (ij < NPOS) {
            const int i = ij / CDIM, j = ij % CDIM;
            int pq = pq0 + row;
            atomicAdd(&D[(i * STRIDE + (pq >> 4)) * IMG + j * STRIDE + (pq & 15)],
                      scale * c0[r]);
            pq = pq0 + 16 + row;
            atomicAdd(&D[(i * STRIDE + (pq >> 4)) * IMG + j * STRIDE + (pq & 15)],
                      scale * c1[r]);
        }
    }
}

// ---------------- encode: R = soft(R + 0.2 * im2col(E) x W^T) ----------------
// grid: (15 ij-tiles, 128 batches), 256 threads = 8 waves.
// Wave w computes the 16(ij) x 16(n) tile n0 = 16w, K = 256 taps.
__global__ __launch_bounds__(256)
void encode_update(const float* __restrict__ E, const float* __restrict__ Wte,
                   float* __restrict__ R, float* __restrict__ acc,
                   const int* __restrict__ flag) {
    if (*flag) return;
    const int ijt = blockIdx.x, b = blockIdx.y;
    const int ij0 = ijt * 16;

    __shared__ float P[16 * PQ];       // 16 KB im2col tile
    __shared__ float red[2][256];
    const float* Eb = E + (size_t)b * IMG * IMG;
    {   // async-stage 64B per thread (one 16-float kernel row of one patch)
        int t = threadIdx.x;
        int r = t >> 4, p = t & 15;    // tile row r, kernel row p
        int ij = ij0 + r;
        if (ij < NPOS) {
            int i = ij / CDIM, j = ij % CDIM;
            async_g2l_b128x4(lds_addr(&P[r * PQ + p * 16]),
                             (uint32_t)(((i * STRIDE + p) * IMG + j * STRIDE) * 4), Eb);
        } else {
            float4 z = make_float4(0.f, 0.f, 0.f, 0.f);
            float4* d = (float4*)&P[r * PQ + p * 16];
            d[0] = z; d[1] = z; d[2] = z; d[3] = z;
        }
    }
    wait_async0();
    __syncthreads();

    const int lane = threadIdx.x & 31;
    const int w    = threadIdx.x >> 5;
    const int row  = lane & 15;
    const int hi   = lane >> 4;
    const int n0   = w * 16;
    const v2f* Wte2 = (const v2f*)Wte;

    v8f c = {};
    for (int k = 0; k < PQ; k += 4) {
        const int kk = k + 2 * hi;
        const int kp = (k >> 1) + hi;
        v2f a  = *(const v2f*)&P[row * PQ + kk];
        v2f bv = Wte2[kp * NATOMS + n0 + row];
        c = wmma4(a, bv, c);
    }

    // SGD + soft threshold + convergence partials
    float dn = 0.0f, dr = 0.0f;
    float* Rb = R + (size_t)b * NPOS_PAD * NATOMS;
    for (int r = 0; r < 8; ++r) {
        const int m  = r + hi * 8;
        const int ij = ij0 + m;
        if (ij < NPOS) {
            const int n   = n0 + row;
            const float o = Rb[ij * NATOMS + n];
            const float pre = o + 2.0f * R_LR * c[r];     // R - lr*(-2*corr)
            const float nw  = fmaxf(pre - LMDA, 0.0f) - fmaxf(-pre - LMDA, 0.0f);
            const float d   = nw - o;
            dn += d * d; dr += o * o;
            Rb[ij * NATOMS + n] = nw;
        }
    }
    red[0][threadIdx.x] = dn; red[1][threadIdx.x] = dr;
    __syncthreads();
    for (int s = 128; s > 0; s >>= 1) {
        if ((int)threadIdx.x < s) {
            red[0][threadIdx.x] += red[0][threadIdx.x + s];
            red[1][threadIdx.x] += red[1][threadIdx.x + s];
        }
        __syncthreads();
    }
    if (threadIdx.x == 0) {
        atomicAdd(&acc[0], red[0][0]);
        atomicAdd(&acc[1], red[1][0]);
    }
}

// ---------------- host-side orchestration ----------------

extern "C" void kernel_launch(void* const* d_in, const int* in_sizes, int n_in,
                              void* d_out, int out_size, void* d_ws, size_t ws_size,
                              hipStream_t stream) {
    const float* img = (const float*)d_in[0];     // [128,1,128,128]
    const float* W   = (const float*)d_in[1];     // [128,1,16,16] -> [n][pq]
    float* out = (float*)d_out;                   // [128,1,128,128]

    float* R   = (float*)d_ws;                                  // [B][240][128]
    float* E   = R   + (size_t)BATCH * NPOS_PAD * NATOMS;       // [B][128][128]
    float* Wd  = E   + (size_t)BATCH * IMG * IMG;               // 64*256 float2
    float* Wte = Wd  + (size_t)2 * 16384;                       // 128*128 float2
    float* acc = Wte + (size_t)2 * 16384;                       // {dn, dr}
    int*  flag = (int*)(acc + 2);

    // init (every call: deterministic, no cross-call state)
    zero_f32<<<2048, 256, 0, stream>>>(R, BATCH * NPOS_PAD * NATOMS);
    pack_W<<<64, 256, 0, stream>>>(W, Wd, Wte);
    init_ctrl<<<1, 1, 0, stream>>>(acc, flag);

    const dim3 grid(CDIM, BATCH);                 // 15 x 128 workgroups
    const int n4 = BATCH * IMG * IMG / 4;

    for (int it = 0; it < MAX_ITER; ++it) {
        copy_img<<<2048, 256, 0, stream>>>((const float4*)img, (float4*)E, n4, flag);
        decode_scatter<<<grid, 256, 0, stream>>>(R, Wd, E, -1.0f, flag); // E = img - convT(R)
        encode_update<<<grid, 256, 0, stream>>>(E, Wte, R, acc, flag);
        check_conv<<<1, 1, 0, stream>>>(acc, flag);
    }

    // final reconstruction: out = convT(R, W)
    zero_f32<<<2048, 256, 0, stream>>>(out, BATCH * IMG * IMG);
    decode_scatter<<<grid, 256, 0, stream>>>(R, Wd, out, 1.0f, nullptr);
}